// _FloatingBaseIKModule_70136815944243
// MI455X (gfx1250) — compile-verified
//
#include <hip/hip_runtime.h>
#include <stdint.h>
#include <stddef.h>

#define NJ   32
#define BLK  256
#define ROWP 33          // padded LDS row stride (dwords): 33 odd -> conflict-free wave32 reads
#define OUTD 94          // 4*6 pose + 32 limit + 32 rest + 6 base

typedef unsigned int u32x4 __attribute__((ext_vector_type(4)));
typedef int          i32x8 __attribute__((ext_vector_type(8)));
typedef int          i32x4 __attribute__((ext_vector_type(4)));

#if defined(__gfx1250__) && __has_builtin(__builtin_amdgcn_tensor_load_to_lds)
#define IK_USE_TDM 1
#else
#define IK_USE_TDM 0
#endif

struct V3 { float x, y, z; };
struct Q4 { float x, y, z, w; };
struct P7 { V3 t; Q4 q; };

__device__ __forceinline__ V3 cross3(V3 a, V3 b) {
  return V3{ a.y*b.z - a.z*b.y, a.z*b.x - a.x*b.z, a.x*b.y - a.y*b.x };
}
__device__ __forceinline__ Q4 qmul(Q4 a, Q4 b) {
  return Q4{ a.w*b.x + b.w*a.x + (a.y*b.z - a.z*b.y),
             a.w*b.y + b.w*a.y + (a.z*b.x - a.x*b.z),
             a.w*b.z + b.w*a.z + (a.x*b.y - a.y*b.x),
             a.w*b.w - (a.x*b.x + a.y*b.y + a.z*b.z) };
}
__device__ __forceinline__ V3 qrot(Q4 q, V3 t) {
  V3 v{ q.x, q.y, q.z };
  V3 u = cross3(v, t);
  u.x += q.w*t.x; u.y += q.w*t.y; u.z += q.w*t.z;
  V3 c = cross3(v, u);
  return V3{ t.x + 2.f*c.x, t.y + 2.f*c.y, t.z + 2.f*c.z };
}
__device__ __forceinline__ P7 pcompose(P7 a, P7 b) {
  V3 r = qrot(a.q, b.t);
  return P7{ V3{ a.t.x + r.x, a.t.y + r.y, a.t.z + r.z }, qmul(a.q, b.q) };
}
__device__ __forceinline__ P7 pinv(P7 T) {
  Q4 qi{ -T.q.x, -T.q.y, -T.q.z, T.q.w };
  V3 r = qrot(qi, T.t);
  return P7{ V3{ -r.x, -r.y, -r.z }, qi };
}
__device__ __forceinline__ void se3log(P7 T, V3& rho, V3& phi) {
  // quat_log (sign-canonicalized)
  float s  = (T.q.w < 0.f) ? -1.f : 1.f;
  float qx = T.q.x*s, qy = T.q.y*s, qz = T.q.z*s, qw = T.q.w*s;
  float nv2 = qx*qx + qy*qy + qz*qz;
  float nv  = sqrtf(fmaxf(nv2, 1e-14f));
  float ang = 2.f*atan2f(nv, qw);
  float wsafe = (fabsf(qw) > 1e-8f) ? qw : 1.f;
  float scale = (nv < 1e-6f) ? (2.f/wsafe) : (ang/nv);
  phi = V3{ qx*scale, qy*scale, qz*scale };
  // V-inverse coefficient
  float th2 = phi.x*phi.x + phi.y*phi.y + phi.z*phi.z;
  float th  = sqrtf(fmaxf(th2, 1e-14f));
  bool  small = th < 1e-4f;
  float ths = small ? 1.f : th;
  float abig = (1.f - ths*sinf(ths)/(2.f*(1.f - cosf(ths)))) / (ths*ths);
  float a = small ? (1.f/12.f + th2/720.f) : abig;
  V3 pt  = cross3(phi, T.t);
  V3 ppt = cross3(phi, pt);
  rho = V3{ T.t.x - 0.5f*pt.x + a*ppt.x,
            T.t.y - 0.5f*pt.y + a*ppt.y,
            T.t.z - 0.5f*pt.z + a*ppt.z };
}

__global__ __launch_bounds__(BLK) void ik_residual_kernel(
    const float* __restrict__ cfg,            // [nb,32]
    const float* __restrict__ base,           // [nb,7]
    const float* __restrict__ target_poses,   // [4,7]
    const float* __restrict__ default_base,   // [7]
    const float* __restrict__ rest_cfg,       // [32]
    const float* __restrict__ joint_offsets,  // [32,7]
    const float* __restrict__ joint_axes,     // [32,3]
    const float* __restrict__ joint_lower,    // [32]
    const float* __restrict__ joint_upper,    // [32]
    float* __restrict__ out,                  // [nb,94]
    int nb)
{
  __shared__ float s_cfg[BLK * ROWP];         // 33792 B, pad-strided by TDM
  const int tid  = threadIdx.x;
  const int blk0 = blockIdx.x * BLK;
  const int b    = blk0 + tid;
  int rows = nb - blk0; if (rows > BLK) rows = BLK;

#if IK_USE_TDM
  if (tid < 32) {                             // one wave issues the DMA
    const uint32_t n  = (uint32_t)rows * NJ;  // elements in this tile (<= 8192)
    const uint64_t ga = (uint64_t)(uintptr_t)(cfg + (size_t)blk0 * NJ);
    const uint32_t ld = (uint32_t)(uintptr_t)&s_cfg[0];   // flat low 32b == LDS offset
    // ---- D# group 0: count=1 | lds_addr | global_addr[56:0] | type=2 ----
    u32x4 g0;
    g0[0] = 1u;
    g0[1] = ld;
    g0[2] = (uint32_t)ga;
    g0[3] = (uint32_t)((ga >> 32) & 0x01FFFFFFull) | (2u << 30);
    // ---- D# group 1: 4B elems, pad every 32 dwords by 1 dword; 1-D tile ----
    i32x8 g1;
    g1[0] = (int)((2u << 16)      // data_size = 4 bytes
                | (1u << 20)      // pad_enable
                | (4u << 22));    // pad_interval: 32 dwords (1<<4 8B units)
                                  // pad_amount encoded 0 -> 1 dword
    g1[1] = (int)(n << 16);                       // tensor_dim0[15:0]
    g1[2] = (int)((n >> 16) | (1u << 16));        // tensor_dim0[31:16] | tensor_dim1=1
    g1[3] = (int)((n & 0xFFFFu) << 16);           // tile_dim0
    g1[4] = 0;                                    // tile_dim1=0 (unused), tile_dim2=0
    g1[5] = (int)n;                               // tensor_dim0_stride[31:0]
    g1[6] = 0;
    g1[7] = 0;
    i32x4 gz = {0, 0, 0, 0};
#if __has_include(<hip/amd_detail/amd_gfx1250_TDM.h>)
    i32x8 gz8 = {0, 0, 0, 0, 0, 0, 0, 0};        // therock/clang-23: 6-arg builtin
    __builtin_amdgcn_tensor_load_to_lds(g0, g1, gz, gz, gz8, 0);
#else
    __builtin_amdgcn_tensor_load_to_lds(g0, g1, gz, gz, 0);  // ROCm 7.2: 5-arg
#endif
    __builtin_amdgcn_s_wait_tensorcnt(0);
  }
#else
  if (b < nb) {
    #pragma unroll
    for (int j = 0; j < NJ; ++j) s_cfg[tid*ROWP + j] = cfg[(size_t)b*NJ + j];
  }
#endif
  __syncthreads();
  if (b >= nb) return;

  // ---- FK scan (serial quaternion chain, all in VGPRs) ----
  P7 Tb0;
  {
    const float* bp = base + (size_t)b * 7;
    Tb0.t = V3{ bp[0], bp[1], bp[2] };
    Tb0.q = Q4{ bp[3], bp[4], bp[5], bp[6] };
  }
  P7 T = Tb0;
  float* o = out + (size_t)b * OUTD;

  #pragma unroll 1
  for (int g = 0; g < 4; ++g) {
    #pragma unroll
    for (int jj = 0; jj < 8; ++jj) {
      const int j = g*8 + jj;                    // wave-uniform -> scalar loads below
      const float* op = joint_offsets + j*7;
      P7 O;
      O.t = V3{ op[0], op[1], op[2] };
      O.q = Q4{ op[3], op[4], op[5], op[6] };
      T = pcompose(T, O);
      const float half = 0.5f * s_cfg[tid*ROWP + j];
      const float sh = sinf(half), ch = cosf(half);
      const float* ap = joint_axes + j*3;
      Q4 jq{ ap[0]*sh, ap[1]*sh, ap[2]*sh, ch };
      T.q = qmul(T.q, jq);
    }
    // pose residual at link (g+1)*8, emitted immediately (no dynamic array)
    const float* tp = target_poses + g*7;
    P7 Tt;
    Tt.t = V3{ tp[0], tp[1], tp[2] };
    Tt.q = Q4{ tp[3], tp[4], tp[5], tp[6] };
    P7 Te = pcompose(pinv(Tt), T);
    V3 rho, phi;
    se3log(Te, rho, phi);
    o[g*6 + 0] = rho.x;       o[g*6 + 1] = rho.y;       o[g*6 + 2] = rho.z;
    o[g*6 + 3] = 0.5f*phi.x;  o[g*6 + 4] = 0.5f*phi.y;  o[g*6 + 5] = 0.5f*phi.z;
  }

  // ---- limit + rest residuals ----
  #pragma unroll
  for (int j = 0; j < NJ; ++j) {
    const float c = s_cfg[tid*ROWP + j];
    o[24 + j] = (fmaxf(c - joint_upper[j], 0.f) + fminf(c - joint_lower[j], 0.f)) * 10.f;
    o[56 + j] = (c - rest_cfg[j]) * 0.1f;
  }

  // ---- base residual ----
  {
    P7 D;
    D.t = V3{ default_base[0], default_base[1], default_base[2] };
    D.q = Q4{ default_base[3], default_base[4], default_base[5], default_base[6] };
    P7 Te = pcompose(pinv(D), Tb0);
    V3 rho, phi;
    se3log(Te, rho, phi);
    o[88] = 5.f*rho.x; o[89] = 5.f*rho.y; o[90] = 5.f*rho.z;
    o[91] = 5.f*phi.x; o[92] = 5.f*phi.y; o[93] = 5.f*phi.z;
  }
}

extern "C" void kernel_launch(void* const* d_in, const int* in_sizes, int n_in,
                              void* d_out, int out_size, void* d_ws, size_t ws_size,
                              hipStream_t stream) {
  const float* cfg  = (const float*)d_in[0];
  const float* base = (const float*)d_in[1];
  const float* tp   = (const float*)d_in[2];
  const float* db   = (const float*)d_in[3];
  const float* rest = (const float*)d_in[4];
  const float* joff = (const float*)d_in[5];
  const float* jaxs = (const float*)d_in[6];
  const float* jlo  = (const float*)d_in[7];
  const float* jup  = (const float*)d_in[8];
  float* out = (float*)d_out;
  const int nb = in_sizes[0] / NJ;
  const int grid = (nb + BLK - 1) / BLK;
  ik_residual_kernel<<<grid, BLK, 0, stream>>>(cfg, base, tp, db, rest, joff,
                                               jaxs, jlo, jup, out, nb);
}